// EGNN_network_2259152798550
// MI455X (gfx1250) — compile-verified
//
#include <hip/hip_runtime.h>
#include <hip/hip_bf16.h>

// ---------------------------------------------------------------------------
// EGNN forward for MI455X (gfx1250, wave32).
//  - Per-edge MLPs as fp32 WMMA (V_WMMA_F32_16X16X4_F32), 22 WMMAs / 16-edge tile,
//    j-loop fully unrolled (8 tiles -> 176 WMMAs) for cross-tile scheduling.
//  - All layer weights live in VGPRs as WMMA B-operands for the whole wave.
//  - All per-lane j-row data (t_j, coors_j) preloaded to registers: zero global
//    loads (and zero loadcnt waits) inside the compute body.
//  - D-layout -> A-layout transposes via per-wave private LDS (in-order DS).
//  - SiLU via v_exp_f32 + v_rcp_f32 hardware transcendentals.
//  - 2 layer launches (grid sync on coors) + tiny 1x1-conv head kernel.
// ---------------------------------------------------------------------------

typedef float v2f __attribute__((ext_vector_type(2)));
typedef float v8f __attribute__((ext_vector_type(8)));

#define B_   64
#define N_   128
#define BN_  (B_ * N_)
#define NT_  (N_ / 16)            // 8 j-tiles per node
#define WPB  8                    // waves per block (256 threads)
#define TS   17                   // padded LDS row stride (bank-conflict relief)
#define WAVE_LDS (16 * TS + 16)   // 16x16 transpose tile + 16 edge weights

// silu(x) = x * sigmoid(x); sigmoid via single v_rcp_f32 (fast, ~1 ulp)
__device__ __forceinline__ float silu_f(float x) {
  return x * __builtin_amdgcn_rcpf(1.0f + __expf(-x));
}
__device__ __forceinline__ float selu_f(float x) {
  const float a = 1.6732632423543772f;
  const float s = 1.0507009873554805f;
  return x > 0.0f ? s * x : s * a * (__expf(x) - 1.0f);
}

// D(16x16,f32) = A(16x4,f32) * B(4x16,f32) + C
__device__ __forceinline__ v8f wmma_k4(v2f a, v2f b, v8f c) {
  return __builtin_amdgcn_wmma_f32_16x16x4_f32(
      /*neg_a=*/false, a, /*neg_b=*/false, b,
      /*c_mod=*/(short)0, c, /*reuse_a=*/false, /*reuse_b=*/false);
}

// One EGNN layer. One wave handles one node (b,i), loops over j in 16-edge tiles.
// A-operand layout (f32 16x4): lane L -> row M = L&15; vgpr r,half h=L>>4 -> K=r+2h.
// B-operand layout (f32 4x16): lane L -> col N = L&15; vgpr r,half h -> K=r+2h.
// C/D layout: vgpr v, lane L -> row M = v + 8*(L>>4), col N = L&15.
__global__ __launch_bounds__(256) void egnn_layer_kernel(
    const float* __restrict__ t, const float* __restrict__ coors_in,
    const float* __restrict__ vel_in,
    const float* __restrict__ We1, const float* __restrict__ be1,
    const float* __restrict__ We2, const float* __restrict__ be2,
    const float* __restrict__ Wc1, const float* __restrict__ bc1,
    const float* __restrict__ Wc2, const float* __restrict__ bc2,
    const float* __restrict__ Wv, const float* __restrict__ bv,
    float* __restrict__ coors_out, float* __restrict__ vel_out) {
  __shared__ float lds[WPB * WAVE_LDS];

  const int lane = threadIdx.x & 31;
  const int wave = threadIdx.x >> 5;
  const int node = blockIdx.x * WPB + wave;        // b*N + i
  const int nb   = (node >> 7) << 7;               // b*N
  const int n    = lane & 15;                      // col / row-in-tile index
  const int h    = lane >> 4;                      // half select
  const int tb   = wave * WAVE_LDS;                // transpose area base
  const int wb   = tb + 16 * TS;                   // per-edge weight area base

  // ---- preload B-operands (resident in VGPRs for the whole wave) ----
  v2f Bw1[2];                                      // We1: (8 x 16), K chunks {0..3},{4..7}
  Bw1[0] = v2f{We1[(0 + 2 * h) * 16 + n], We1[(1 + 2 * h) * 16 + n]};
  Bw1[1] = v2f{We1[(4 + 2 * h) * 16 + n], We1[(5 + 2 * h) * 16 + n]};
  v2f Bw2[4];                                      // We2: (16 x 16)
#pragma unroll
  for (int c = 0; c < 4; ++c)
    Bw2[c] = v2f{We2[(4 * c + 2 * h) * 16 + n], We2[(4 * c + 2 * h + 1) * 16 + n]};
  v2f BwC[4][4];                                   // Wc1: (16 x 64), 4 N-tiles
#pragma unroll
  for (int nt = 0; nt < 4; ++nt)
#pragma unroll
    for (int c = 0; c < 4; ++c)
      BwC[nt][c] = v2f{Wc1[(4 * c + 2 * h) * 64 + 16 * nt + n],
                       Wc1[(4 * c + 2 * h + 1) * 64 + 16 * nt + n]};

  const float be1l = be1[n];
  const float be2l = be2[n];
  float bc1l[4], wc2l[4];
#pragma unroll
  for (int nt = 0; nt < 4; ++nt) { bc1l[nt] = bc1[16 * nt + n]; wc2l[nt] = Wc2[16 * nt + n]; }
  const float bc2s = bc2[0];
  const float phiW = Wv[0] + Wv[1] + Wv[2];        // feats = t repeated 3x
  const float bvs  = bv[0];

  // ---- node state ----
  const float ti  = t[node];
  const float cix = coors_in[2 * node + 0];
  const float ciy = coors_in[2 * node + 1];
  const float vix = vel_in[2 * node + 0];
  const float viy = vel_in[2 * node + 1];

  // ---- preload ALL j-row data for this lane (8 tiles x {t_j, coor_j}) ----
  float tj[NT_], cjx[NT_], cjy[NT_];
#pragma unroll
  for (int jt = 0; jt < NT_; ++jt) {
    const int jidx = nb + jt * 16 + n;
    tj[jt]  = t[jidx];
    cjx[jt] = coors_in[2 * jidx + 0];
    cjy[jt] = coors_in[2 * jidx + 1];
  }

  float aggx = 0.0f, aggy = 0.0f;

#pragma unroll
  for (int jt = 0; jt < NT_; ++jt) {
    const float rlx = cix - cjx[jt];
    const float rly = ciy - cjy[jt];
    const float dst = rlx * rlx + rly * rly;

    // e_in K order: [hi0..2 = ti, hj0..2 = tj, dist, edge=1]
    v2f A0, A1;
    A0[0] = ti;                  A0[1] = h ? tj[jt] : ti;      // K {0|2}, {1|3}
    A1[0] = h ? dst : tj[jt];    A1[1] = h ? 1.0f : tj[jt];    // K {4|6}, {5|7}

    // --- edge MLP layer 1: (16x8)@(8x16) ---
    v8f acc = {};
    acc = wmma_k4(A0, Bw1[0], acc);
    acc = wmma_k4(A1, Bw1[1], acc);
#pragma unroll
    for (int v = 0; v < 8; ++v)
      lds[tb + (v + 8 * h) * TS + n] = silu_f(acc[v] + be1l);   // D->A transpose
    v2f A2[4];
#pragma unroll
    for (int c = 0; c < 4; ++c) {
      const int k0 = 4 * c + 2 * h;
      A2[c] = v2f{lds[tb + n * TS + k0], lds[tb + n * TS + k0 + 1]};
    }

    // --- edge MLP layer 2: (16x16)@(16x16) -> m ---
    v8f acc2 = {};
#pragma unroll
    for (int c = 0; c < 4; ++c) acc2 = wmma_k4(A2[c], Bw2[c], acc2);
#pragma unroll
    for (int v = 0; v < 8; ++v)
      lds[tb + (v + 8 * h) * TS + n] = silu_f(silu_f(acc2[v] + be2l));  // silu(m)
    v2f A3[4];
#pragma unroll
    for (int c = 0; c < 4; ++c) {
      const int k0 = 4 * c + 2 * h;
      A3[c] = v2f{lds[tb + n * TS + k0], lds[tb + n * TS + k0 + 1]};
    }

    // --- coors MLP: (16x16)@(16x64), fold Wc2 (64->1) into D-layout ---
    float s[8] = {0, 0, 0, 0, 0, 0, 0, 0};
#pragma unroll
    for (int nt = 0; nt < 4; ++nt) {
      v8f hc = {};
#pragma unroll
      for (int c = 0; c < 4; ++c) hc = wmma_k4(A3[c], BwC[nt][c], hc);
#pragma unroll
      for (int v = 0; v < 8; ++v)
        s[v] += silu_f(hc[v] + bc1l[nt]) * wc2l[nt];
    }
    // reduce over the 16 columns: xor masks 1,2,4,8 stay within each half
#pragma unroll
    for (int m = 1; m < 16; m <<= 1)
#pragma unroll
      for (int v = 0; v < 8; ++v) s[v] += __shfl_xor(s[v], m, 32);

    // scatter per-edge weights: lane0 -> edges 0..7, lane16 -> edges 8..15
    if (n == 0) {
#pragma unroll
      for (int v = 0; v < 8; ++v) lds[wb + 8 * h + v] = s[v];
    }
    const float wj   = lds[wb + n] + bc2s;
    const float gate = (lane < 16) ? 1.0f : 0.0f;  // lanes>=16 hold duplicate j's
    aggx += gate * wj * rlx;
    aggy += gate * wj * rly;
  }

  // reduce agg across all 32 lanes (upper half contributed zero)
#pragma unroll
  for (int m = 1; m < 32; m <<= 1) {
    aggx += __shfl_xor(aggx, m, 32);
    aggy += __shfl_xor(aggy, m, 32);
  }

  const float phi = ti * phiW + bvs;
  float nvx = phi * vix + aggx;
  float nvy = phi * viy + aggy;
  float ncx = cix + nvx;
  float ncy = ciy + nvy;
  ncx = selu_f(ncx); ncy = selu_f(ncy);
  nvx = selu_f(nvx); nvy = selu_f(nvy);

  if (lane == 0) {
    coors_out[2 * node + 0] = ncx;
    coors_out[2 * node + 1] = ncy;
    vel_out[2 * node + 0]   = nvx;
    vel_out[2 * node + 1]   = nvy;
  }
}

// 1x1-conv head: 4 -> 32 -> 4 channels per node; trivial VALU work.
__global__ __launch_bounds__(256) void egnn_head_kernel(
    const float* __restrict__ coors, const float* __restrict__ vel,
    const float* __restrict__ W1, const float* __restrict__ b1,
    const float* __restrict__ W2, const float* __restrict__ b2,
    float* __restrict__ out) {
  const int i = blockIdx.x * blockDim.x + threadIdx.x;
  if (i >= BN_) return;
  float x[4] = {selu_f(coors[2 * i]), selu_f(coors[2 * i + 1]),
                selu_f(vel[2 * i]),   selu_f(vel[2 * i + 1])};
  float hbuf[32];
#pragma unroll
  for (int o = 0; o < 32; ++o) {
    float a = b1[o];
#pragma unroll
    for (int c = 0; c < 4; ++c) a += x[c] * W1[o * 4 + c];
    hbuf[o] = selu_f(a);
  }
  float y[4];
#pragma unroll
  for (int o = 0; o < 4; ++o) {
    float a = b2[o];
#pragma unroll
    for (int c = 0; c < 32; ++c) a += hbuf[c] * W2[o * 32 + c];
    y[o] = a;
  }
  out[2 * i + 0] = y[0];
  out[2 * i + 1] = y[1];
  out[2 * BN_ + 2 * i + 0] = y[2];
  out[2 * BN_ + 2 * i + 1] = y[3];
}

extern "C" void kernel_launch(void* const* d_in, const int* in_sizes, int n_in,
                              void* d_out, int out_size, void* d_ws, size_t ws_size,
                              hipStream_t stream) {
  const float* t      = (const float*)d_in[0];
  const float* coors  = (const float*)d_in[1];
  const float* vel    = (const float*)d_in[2];
  const float* We1    = (const float*)d_in[3];
  const float* be1    = (const float*)d_in[4];
  const float* We2    = (const float*)d_in[5];
  const float* be2    = (const float*)d_in[6];
  const float* Wc1    = (const float*)d_in[7];
  const float* bc1    = (const float*)d_in[8];
  const float* Wc2    = (const float*)d_in[9];
  const float* bc2    = (const float*)d_in[10];
  const float* Wv     = (const float*)d_in[11];
  const float* bv     = (const float*)d_in[12];
  const float* Wconv1 = (const float*)d_in[13];
  const float* bconv1 = (const float*)d_in[14];
  const float* Wconv2 = (const float*)d_in[15];
  const float* bconv2 = (const float*)d_in[16];

  float* ws = (float*)d_ws;
  float* c0 = ws;             // 16384 floats each (B*N*2)
  float* v0 = ws + 16384;
  float* c1 = ws + 32768;
  float* v1 = ws + 49152;

  const int layer_blocks = BN_ / WPB;  // 1024 blocks x 256 threads, 1 wave per node

  // layer 0
  egnn_layer_kernel<<<layer_blocks, 256, 0, stream>>>(
      t, coors, vel,
      We1, be1, We2, be2, Wc1, bc1, Wc2, bc2, Wv, bv,
      c0, v0);
  // layer 1 (per-layer weight slices)
  egnn_layer_kernel<<<layer_blocks, 256, 0, stream>>>(
      t, c0, v0,
      We1 + 128, be1 + 16, We2 + 256, be2 + 16,
      Wc1 + 1024, bc1 + 64, Wc2 + 64, bc2 + 1, Wv + 3, bv + 1,
      c1, v1);
  // conv head
  egnn_head_kernel<<<BN_ / 256, 256, 0, stream>>>(
      c1, v1, Wconv1, bconv1, Wconv2, bconv2, (float*)d_out);
}